// LocalExpansion_83597243449798
// MI455X (gfx1250) — compile-verified
//
#include <hip/hip_runtime.h>
#include <cstdint>

// LocalExpansion: out[b,h,y,x,k,d] = in[b,h,y+i-3,x+j-3,d], k = i*7+j, zero-padded.
// Zero-FLOP gather: 9.4 MB in, 462 MB out -> pure store-bound (~20us @ 23.3 TB/s).
// Strategy: per (b*h, y) block, async-copy the 7 needed input rows (zero-padded
// tile) into LDS once (gfx1250 ASYNCcnt path), then serve the 49x reuse from LDS
// with ds_load_b128 and write perfectly contiguous global_store_b128.

#define KH   7
#define KW   7
#define PAD  3
#define HT   48
#define WD   48
#define DIM  64
#define KK   (KH * KW)            // 49
#define ROWS (1 + 2 * PAD)        // 7 staged input rows per output row
#define WPAD (WD + 2 * PAD)       // 54 padded columns
#define LDS_FLOATS (ROWS * WPAD * DIM)   // 24192 floats = 94.5 KB -> 3 blocks/WGP
#define NBH    16                 // B * N_HEADS
#define THREADS 256

__global__ __launch_bounds__(THREADS)
void local_expansion_kernel(const float* __restrict__ x, float* __restrict__ out) {
    extern __shared__ float smem[];          // ROWS x WPAD x DIM
    float4* s4 = (float4*)smem;

    const int t   = threadIdx.x;
    const int blk = blockIdx.x;
    const int y   = blk % HT;                // output row
    const int bh  = blk / HT;                // fused batch*head

    // ---- 1) zero the padded tile (boundary rows/cols must read as 0) ----
    const float4 z = make_float4(0.f, 0.f, 0.f, 0.f);
    for (int i = t; i < LDS_FLOATS / 4; i += THREADS) s4[i] = z;
    __syncthreads();

    // ---- 2) async-copy valid input rows global -> LDS (gfx1250 ASYNCcnt path) ----
    const float*   src      = x + (size_t)bh * (HT * WD * DIM);
    const uint32_t lds_base = (uint32_t)(uintptr_t)smem;  // LDS aperture: offset in addr[31:0]
    for (int r = 0; r < ROWS; ++r) {
        const int gy = y - PAD + r;                        // uniform across block
        if (gy < 0 || gy >= HT) continue;                  // stays zero
        // one row = 48*64 floats = 768 float4; 256 threads -> 3 each
#pragma unroll
        for (int c = 0; c < 3; ++c) {
            const int f4  = t + c * THREADS;               // 0..767
            const int col = f4 >> 4;                       // 0..47
            const int dch = f4 & 15;                       // float4 within d=64
            const uint32_t goff = (uint32_t)(gy * WD * DIM + f4 * 4) * 4u;
            const uint32_t loff = lds_base +
                ((uint32_t)((r * WPAD + (col + PAD)) * DIM + dch * 4)) * 4u;
            asm volatile("global_load_async_to_lds_b128 %0, %1, %2"
                         :: "v"(loff), "v"(goff), "s"(src)
                         : "memory");
        }
    }
    asm volatile("s_wait_asynccnt 0" ::: "memory");
    __syncthreads();

    // ---- 3) expand: 49 shifted copies per pixel ----
    // Output float4 index within this row = ((x*49 + k)*64 + d)/4 = pair*16 + lane16
    //                                     = it*256 + t  -> contiguous flat store.
    float4* dst4 = (float4*)out +
                   ((size_t)bh * (HT * WD) + (size_t)y * WD) * KK * (DIM / 4);
    const int lane16 = t & 15;      // which float4 of d=64
    const int pairo  = t >> 4;      // 16 (x,k) pairs per iteration -> 4KB contiguous

#pragma unroll 3
    for (int it = 0; it < (WD * KK) / 16; ++it) {          // 147 iterations
        const int pair = it * 16 + pairo;                  // 0..2351 = x*49 + k
        const int xcol = pair / KK;                        // 0..47
        const int k    = pair - xcol * KK;                 // 0..48
        const int i    = k / KW;                           // 0..6  (LDS row)
        const int j    = k - i * KW;
        const float4 v = s4[(i * WPAD + (xcol + j)) * 16 + lane16];
        dst4[it * THREADS + t] = v;
    }
}

extern "C" void kernel_launch(void* const* d_in, const int* in_sizes, int n_in,
                              void* d_out, int out_size, void* d_ws, size_t ws_size,
                              hipStream_t stream) {
    (void)in_sizes; (void)n_in; (void)d_ws; (void)ws_size; (void)out_size;
    const float* x = (const float*)d_in[0];   // [2,8,2304,64] fp32; height/width scalars ignored (48x48)
    float* out     = (float*)d_out;           // [2,8,2304,49,64] fp32
    const dim3 grid(NBH * HT);                // 768 blocks
    const dim3 block(THREADS);
    const size_t smem_bytes = (size_t)LDS_FLOATS * sizeof(float);  // 96768 B
    local_expansion_kernel<<<grid, block, smem_bytes, stream>>>(x, out);
}